// RegionalCosineSimilarityLoss_67181878444469
// MI455X (gfx1250) — compile-verified
//
#include <hip/hip_runtime.h>
#include <math.h>

// Problem constants (match the reference).
#define B_DIM 8
#define T_DIM 2048
#define C_DIM 256
#define NBHD  2
#define TILE  16
#define NTILES (T_DIM / TILE)   // 128

typedef __attribute__((ext_vector_type(2))) float v2f;
typedef __attribute__((ext_vector_type(8))) float v8f;

// ---------------------------------------------------------------------------
// Kernel 0: zero the two global accumulators (sum, count).
// ---------------------------------------------------------------------------
__global__ void rcs_zero_kernel(float* accum) {
    if (threadIdx.x < 2) accum[threadIdx.x] = 0.0f;
}

// ---------------------------------------------------------------------------
// Kernel 1: row-normalize x (B*T rows of C=256 floats) into xn.
// One wave32 per row; each lane owns 8 contiguous floats (two float4 loads),
// sum-of-squares reduced across the wave with shfl_xor.
// ---------------------------------------------------------------------------
__global__ void rcs_normalize_kernel(const float* __restrict__ x,
                                     float* __restrict__ xn) {
    const int wave = threadIdx.x >> 5;              // 0..7
    const int lane = threadIdx.x & 31;
    const int row  = blockIdx.x * 8 + wave;         // < B*T
    const size_t base = (size_t)row * C_DIM + (size_t)lane * 8;

    float4 v0 = *(const float4*)(x + base);
    float4 v1 = *(const float4*)(x + base + 4);

    float s = v0.x * v0.x + v0.y * v0.y + v0.z * v0.z + v0.w * v0.w +
              v1.x * v1.x + v1.y * v1.y + v1.z * v1.z + v1.w * v1.w;
    #pragma unroll
    for (int off = 16; off > 0; off >>= 1)
        s += __shfl_xor(s, off, 32);

    const float norm  = sqrtf(s);
    const float scale = 1.0f / fmaxf(norm, 1e-8f);

    v0.x *= scale; v0.y *= scale; v0.z *= scale; v0.w *= scale;
    v1.x *= scale; v1.y *= scale; v1.z *= scale; v1.w *= scale;
    *(float4*)(xn + base)     = v0;
    *(float4*)(xn + base + 4) = v1;
}

// ---------------------------------------------------------------------------
// Kernel 2: one wave32 per 16x16 output tile of cos_sim = Xn * Xn^T.
// Accumulates masked, signed similarities + pair count into global accums.
//
// f32 WMMA fragment layouts (CDNA5 ISA 7.12.2):
//   A (16x4):  lane L<16 holds (M=L, K={k,k+1}); lane L>=16 holds (M=L-16, K={k+2,k+3})
//   B (4x16):  lane L<16 holds (N=L, K={k,k+1}); lane L>=16 holds (N=L-16, K={k+2,k+3})
//   C/D (16x16): VGPR v holds M = v + 8*(lane>=16), N = lane&15
// Since both operands come from the same normalized matrix, A and B loads are
// identical float2 gathers from different row bases.
// ---------------------------------------------------------------------------
__global__ void rcs_tile_kernel(const float* __restrict__ xn,
                                const int* __restrict__ regions,
                                float* __restrict__ accum) {
    const int ti = blockIdx.x;
    const int tj = blockIdx.y;
    const int b  = blockIdx.z;
    if (tj < ti) return;                            // triu(k=1): only j >= i tiles

    const int* regB = regions + (size_t)b * T_DIM;

    // Regions are sorted along T: if the smallest region in the column tile is
    // more than NBHD above the largest region in the row tile, the whole tile
    // is masked out. (Uniform scalar branch -> safe before WMMA.)
    const int rowMax = regB[ti * TILE + (TILE - 1)];
    const int colMin = regB[tj * TILE];
    if (colMin - rowMax > NBHD) return;

    const int lane = threadIdx.x;                   // 0..31, all active
    const int mn   = lane & 15;
    const int hi   = lane >> 4;
    const int koff = hi * 2;

    const float* arow = xn + ((size_t)(b * T_DIM + ti * TILE + mn)) * C_DIM + koff;
    const float* brow = xn + ((size_t)(b * T_DIM + tj * TILE + mn)) * C_DIM + koff;

    v8f c = {};
    #pragma unroll 8
    for (int k = 0; k < C_DIM; k += 4) {
        v2f a  = *(const v2f*)(arow + k);
        v2f bb = *(const v2f*)(brow + k);
        // D = A * B + C ; emits v_wmma_f32_16x16x4_f32
        c = __builtin_amdgcn_wmma_f32_16x16x4_f32(
                /*neg_a=*/false, a, /*neg_b=*/false, bb,
                /*c_mod=*/(short)0, c, /*reuse_a=*/false, /*reuse_b=*/false);
    }

    // Apply mask/sign per element of the C fragment and reduce.
    const int   s_idx = tj * TILE + mn;
    const int   rs    = regB[s_idx];
    float sum = 0.0f, cnt = 0.0f;
    #pragma unroll
    for (int v = 0; v < 8; ++v) {
        const int t_idx = ti * TILE + v + 8 * hi;
        const int rt    = regB[t_idx];
        const int d     = (rt > rs) ? (rt - rs) : (rs - rt);
        const bool ok   = (s_idx > t_idx) & (d <= NBHD) & (rt != 0) & (rs != 0);
        if (ok) {
            sum += ((d == 0) ? 1.0f : -1.0f) * c[v];
            cnt += 1.0f;
        }
    }
    #pragma unroll
    for (int off = 16; off > 0; off >>= 1) {
        sum += __shfl_xor(sum, off, 32);
        cnt += __shfl_xor(cnt, off, 32);
    }
    if (lane == 0 && cnt != 0.0f) {
        atomicAdd(accum + 0, sum);
        atomicAdd(accum + 1, cnt);
    }
}

// ---------------------------------------------------------------------------
// Kernel 3: finalize the scalar loss.
// ---------------------------------------------------------------------------
__global__ void rcs_finalize_kernel(const float* __restrict__ accum,
                                    float* __restrict__ out) {
    if (threadIdx.x == 0)
        out[0] = 1.0f - accum[0] / (accum[1] + 1e-6f);
}

// ---------------------------------------------------------------------------
// Launch: x -> normalize -> banded tiled WMMA reduction -> scalar.
// Workspace layout: [0, 16MB) normalized x, then 2 accumulator floats.
// ---------------------------------------------------------------------------
extern "C" void kernel_launch(void* const* d_in, const int* in_sizes, int n_in,
                              void* d_out, int out_size, void* d_ws, size_t ws_size,
                              hipStream_t stream) {
    (void)in_sizes; (void)n_in; (void)out_size;

    const float* x       = (const float*)d_in[0];
    const int*   regions = (const int*)d_in[1];
    float*       loss    = (float*)d_out;

    const size_t xn_elems = (size_t)B_DIM * T_DIM * C_DIM;
    float* xn    = (float*)d_ws;
    float* accum = (float*)((char*)d_ws + xn_elems * sizeof(float));
    (void)ws_size;  // requires xn_elems*4 + 8 bytes (~16.8 MB)

    rcs_zero_kernel<<<1, 32, 0, stream>>>(accum);

    // 8 rows per 256-thread block (8 waves), one wave per row.
    rcs_normalize_kernel<<<(B_DIM * T_DIM) / 8, 256, 0, stream>>>(x, xn);

    // One wave32 per 16x16 tile; region-band pruning inside.
    dim3 grid(NTILES, NTILES, B_DIM);
    rcs_tile_kernel<<<grid, 32, 0, stream>>>(xn, regions, accum);

    rcs_finalize_kernel<<<1, 32, 0, stream>>>(accum, loss);
}